// RGCN_59210419143214
// MI455X (gfx1250) — compile-verified
//
#include <hip/hip_runtime.h>
#include <hip/hip_bf16.h>

typedef __attribute__((ext_vector_type(2))) float v2f;
typedef __attribute__((ext_vector_type(8))) float v8f;

#define NRELS 24
#define HD 64
#define TPW 4                      // tiles (16 edges each) per wave
#define WPACK_ELTS (NRELS * 4 * 16 * 32)   // v2f elements per packed W

// ---------------- binning: histogram + degree ----------------
__global__ void count_deg_kernel(const int* __restrict__ rel, const int* __restrict__ dst,
                                 int E, int* __restrict__ cnt, float* __restrict__ deg) {
    int e = blockIdx.x * blockDim.x + threadIdx.x;
    if (e < E) {
        atomicAdd(&cnt[rel[e]], 1);
        atomicAdd(&deg[dst[e]], 1.0f);
    }
}

// exclusive scan over 24 relation counts, each padded to 16-edge tiles
__global__ void scan_kernel(const int* __restrict__ cnt, int* __restrict__ poff,
                            int* __restrict__ run) {
    if (threadIdx.x == 0) {
        int acc = 0;
        for (int r = 0; r < NRELS; ++r) {
            poff[r] = acc;
            acc += (cnt[r] + 15) & ~15;
            run[r] = 0;
        }
        poff[NRELS] = acc;
    }
}

__global__ void perm_kernel(const int* __restrict__ rel, int E, const int* __restrict__ poff,
                            int* __restrict__ run, int* __restrict__ perm) {
    int e = blockIdx.x * blockDim.x + threadIdx.x;
    if (e < E) {
        int r = rel[e];
        int p = atomicAdd(&run[r], 1);
        perm[poff[r] + p] = e;
    }
}

// ---------------- W repack into WMMA B-operand lane order ----------------
// wp[((r*4+nt)*16+kk)*32 + lane] = { W[r][k0][col], W[r][k0+1][col] }
// with lane = kh*16+row, k0 = kk*4+kh*2, col = nt*16+row.
// Main kernel then fetches B for one WMMA as a single coalesced b64 per lane.
__global__ void wpack_kernel(const float* __restrict__ W, v2f* __restrict__ wp) {
    int i = blockIdx.x * blockDim.x + threadIdx.x;
    if (i >= WPACK_ELTS) return;
    int lane = i & 31;
    int kk   = (i >> 5) & 15;
    int nt   = (i >> 9) & 3;
    int r    = i >> 11;
    int row = lane & 15, kh = lane >> 4;
    int k0 = kk * 4 + kh * 2;
    const float* wr = W + (size_t)r * HD * HD;
    v2f v;
    v.x = wr[(size_t)(k0    ) * HD + nt * 16 + row];
    v.y = wr[(size_t)(k0 + 1) * HD + nt * 16 + row];
    wp[i] = v;
}

// ---------------- core: per-relation tiled GEMM via f32 WMMA ----------------
// One wave32 per TPW consecutive 16-edge tiles (same relation except at the
// <=23 segment boundaries). B (all 64 v2f operands of W[r]) stays resident in
// VGPRs across tiles; A is gathered per tile; C scatter-added with f32 atomics
// into the L2-resident agg buffer.
__global__ void __launch_bounds__(32)
rgcn_wmma_kernel(const float* __restrict__ hin, const int* __restrict__ src,
                 const int* __restrict__ dst, const int* __restrict__ perm,
                 const int* __restrict__ poff, const v2f* __restrict__ wp,
                 float* __restrict__ agg) {
    const int lane  = threadIdx.x;
    const int total = poff[NRELS];
    const int row = lane & 15;
    const int kh  = lane >> 4;

    v2f B[64];          // resident B operands: index nt*16+kk
    int rcur = -1;

    for (int tg = 0; tg < TPW; ++tg) {
        const int base = (blockIdx.x * TPW + tg) * 16;
        if (base >= total) break;                 // uniform exit

        int r = 0;
#pragma unroll
        for (int i = 1; i < NRELS; ++i) r = (base >= poff[i]) ? i : r;

        if (r != rcur) {                          // uniform branch
            const v2f* w = wp + (size_t)r * 64 * 32 + lane;
#pragma unroll
            for (int q = 0; q < 64; ++q) B[q] = w[(size_t)q * 32];
            rcur = r;
        }

        // A-matrix gather: lane holds K = kk*4 + kh*2 + {0,1} of row (lane&15)
        const int e    = perm[base + row];
        const int srow = (e >= 0) ? src[e] : 0;
        const float fv = (e >= 0) ? 1.0f : 0.0f;
        const float* hp = hin + (size_t)srow * HD + kh * 2;
        v2f a[16];
#pragma unroll
        for (int kk = 0; kk < 16; ++kk) {
            v2f av = *(const v2f*)(hp + kk * 4);
            av.x *= fv;                           // sentinel rows -> zero
            av.y *= fv;
            a[kk] = av;
        }

        v8f c[4];
#pragma unroll
        for (int nt = 0; nt < 4; ++nt) c[nt] = (v8f){0.f,0.f,0.f,0.f,0.f,0.f,0.f,0.f};
#pragma unroll
        for (int nt = 0; nt < 4; ++nt) {
#pragma unroll
            for (int kk = 0; kk < 16; ++kk) {
                c[nt] = __builtin_amdgcn_wmma_f32_16x16x4_f32(
                    false, a[kk], false, B[nt * 16 + kk], (short)0, c[nt], false, false);
            }
        }

        // C/D layout: VGPR j of lane l = (M = j + 8*(l>>4), N = l&15)
#pragma unroll
        for (int j = 0; j < 8; ++j) {
            const int m  = kh * 8 + j;
            const int em = perm[base + m];
            if (em >= 0) {
                float* ap = agg + (size_t)dst[em] * HD;
#pragma unroll
                for (int nt = 0; nt < 4; ++nt)
                    atomicAdd(&ap[nt * 16 + row], c[nt][j]);
            }
        }
    }
}

// ---------------- mean + relu ----------------
__global__ void finalize_kernel(const float* __restrict__ agg, const float* __restrict__ deg,
                                float* __restrict__ out, int total) {
    int i = blockIdx.x * blockDim.x + threadIdx.x;
    if (i < total) {
        float d = deg[i >> 6];
        float v = agg[i] / fmaxf(d, 1.0f);
        out[i] = fmaxf(v, 0.0f);
    }
}

extern "C" void kernel_launch(void* const* d_in, const int* in_sizes, int n_in,
                              void* d_out, int out_size, void* d_ws, size_t ws_size,
                              hipStream_t stream) {
    const float* h   = (const float*)d_in[0];
    const int*   src = (const int*)d_in[1];
    const int*   dst = (const int*)d_in[2];
    const int*   rel = (const int*)d_in[3];
    const float* W1  = (const float*)d_in[4];
    const float* W2  = (const float*)d_in[5];
    const int N = in_sizes[0] / HD;
    const int E = in_sizes[1];

    // workspace carve
    int*   cnt  = (int*)d_ws;                       // 32 ints
    int*   poff = cnt + 32;                         // 32 ints
    int*   run  = poff + 32;                        // 32 ints
    float* deg  = (float*)(run + 32);               // N floats
    int*   perm = (int*)(deg + N);                  // E + NRELS*16 ints
    float* agg  = (float*)(perm + E + NRELS * 16);  // N*HD floats
    v2f*   wp1  = (v2f*)(agg + (size_t)N * HD);     // WPACK_ELTS v2f
    v2f*   wp2  = wp1 + WPACK_ELTS;                 // WPACK_ELTS v2f
    float* out  = (float*)d_out;

    hipMemsetAsync(cnt, 0, 3 * 32 * sizeof(int), stream);
    hipMemsetAsync(deg, 0, (size_t)N * sizeof(float), stream);
    hipMemsetAsync(perm, 0xFF, (size_t)(E + NRELS * 16) * sizeof(int), stream);
    hipMemsetAsync(agg, 0, (size_t)N * HD * sizeof(float), stream);

    count_deg_kernel<<<(E + 255) / 256, 256, 0, stream>>>(rel, dst, E, cnt, deg);
    scan_kernel<<<1, 32, 0, stream>>>(cnt, poff, run);
    perm_kernel<<<(E + 255) / 256, 256, 0, stream>>>(rel, E, poff, run, perm);
    wpack_kernel<<<(WPACK_ELTS + 255) / 256, 256, 0, stream>>>(W1, wp1);
    wpack_kernel<<<(WPACK_ELTS + 255) / 256, 256, 0, stream>>>(W2, wp2);

    const int maxTiles = (E + NRELS * 16 + 15) / 16;
    const int nBlocks  = (maxTiles + TPW - 1) / TPW;
    const int ftotal   = N * HD;

    // layer 1: h -> agg -> out (acts as h1)
    rgcn_wmma_kernel<<<nBlocks, 32, 0, stream>>>(h, src, dst, perm, poff, wp1, agg);
    finalize_kernel<<<(ftotal + 255) / 256, 256, 0, stream>>>(agg, deg, out, ftotal);

    // layer 2: out (h1) -> agg -> out
    hipMemsetAsync(agg, 0, (size_t)N * HD * sizeof(float), stream);
    rgcn_wmma_kernel<<<nBlocks, 32, 0, stream>>>(out, src, dst, perm, poff, wp2, agg);
    finalize_kernel<<<(ftotal + 255) / 256, 256, 0, stream>>>(agg, deg, out, ftotal);
}